// RobustCombiner_49280454754798
// MI455X (gfx1250) — compile-verified
//
#include <hip/hip_runtime.h>
#include <hip/hip_bf16.h>

#define TOK 2048
#define KNN 32
#define VOC 32000

typedef __attribute__((ext_vector_type(2))) float v2f;
typedef __attribute__((ext_vector_type(4))) float v4f;
typedef __attribute__((ext_vector_type(8))) float v8f;

// ---------------- Kernel Z: zero knn_prob region (262 MB, streaming) -------
__global__ __launch_bounds__(256) void zero_kernel(float* __restrict__ out, size_t n4) {
    size_t i = (size_t)blockIdx.x * blockDim.x + threadIdx.x;
    size_t stride = (size_t)gridDim.x * blockDim.x;
    v4f z = {0.f, 0.f, 0.f, 0.f};
    for (; i < n4; i += stride)
        __builtin_nontemporal_store(z, ((v4f*)out) + i);
}

// ---------------- Kernel T: per-token wave32: top-8 scan + logs + counts ---
__global__ __launch_bounds__(256) void token_kernel(
    const int*   __restrict__ tgt,
    const float* __restrict__ dists,
    const float* __restrict__ keyf,
    const float* __restrict__ nprobs,
    const float* __restrict__ selp,
    const float* __restrict__ Wf,  const float* __restrict__ bf,
    const float* __restrict__ W1a, const float* __restrict__ b1a,
    const float* __restrict__ W1b, const float* __restrict__ b1b,
    float* __restrict__ feat,      // [TOK][64]  (dists | counts)
    float* __restrict__ noise,     // [TOK][32]
    float* __restrict__ simlam)    // [TOK]
{
    const int lane = threadIdx.x & 31;
    const int tok  = blockIdx.x * 8 + (threadIdx.x >> 5);

    // ---- streaming top-8 (per-lane sorted-descending register array) ----
    const v4f* np4 = (const v4f*)(nprobs + (size_t)tok * VOC);
    float t[8];
#pragma unroll
    for (int i = 0; i < 8; ++i) t[i] = -1.0f;

    for (int it = 0; it < VOC / 128; ++it) {           // 250 iters, b128 loads
        v4f v4 = __builtin_nontemporal_load(np4 + (size_t)it * 32 + lane);
#pragma unroll
        for (int c = 0; c < 4; ++c) {
            float v = v4[c];
            if (v > t[7]) {
                t[7] = v;
#pragma unroll
                for (int j = 7; j >= 1; --j) {
                    if (t[j] > t[j-1]) { float tmp = t[j-1]; t[j-1] = t[j]; t[j] = tmp; }
                }
            }
        }
    }

    // ---- 8-round wave argmax merge; accumulate W_func . log(top8) ----
    float simacc = 0.0f;   // identical on all lanes
#pragma unroll
    for (int r = 0; r < 8; ++r) {
        float v = t[0];
        int   s = lane;
#pragma unroll
        for (int off = 16; off >= 1; off >>= 1) {
            float ov = __shfl_xor(v, off);
            int   os = __shfl_xor(s, off);
            if (ov > v || (ov == v && os < s)) { v = ov; s = os; }
        }
        simacc += Wf[r] * logf(v);
        if (lane == s) {            // winner pops its head
#pragma unroll
            for (int j = 0; j < 7; ++j) t[j] = t[j+1];
            t[7] = -1.0f;
        }
    }

    // ---- per-(token,k) features ----
    const size_t base = (size_t)tok * KNN;
    float dist = dists[base + lane];
    float lk   = logf(keyf[base + lane]);
    float ls   = logf(selp[base + lane]);

    // noise MLP: 2 -> 4 tanh -> 1
    float nz = b1b[0];
#pragma unroll
    for (int j = 0; j < 4; ++j)
        nz += W1b[j] * tanhf(W1a[2*j] * lk + W1a[2*j + 1] * ls + b1a[j]);

    // per-lane part of sim_lambda dot
    float part = Wf[8 + lane] * lk + Wf[40 + lane] * ls;
#pragma unroll
    for (int off = 16; off >= 1; off >>= 1) part += __shfl_xor(part, off);

    // distinct-nonzero-label prefix counts + first-occurrence bookkeeping
    int lbl = tgt[base + lane];
    bool dupB = false;
    for (int j = 0; j < 32; ++j) {
        int lj = __shfl(lbl, j);
        if (j < lane && lj == lbl) dupB = true;
    }
    unsigned mask = (unsigned)__ballot((lbl != 0) && !dupB);
    int cnt = __popc(mask & ((2u << lane) - 1u));      // inclusive prefix

    feat[(size_t)tok * 64 + lane]      = dist;
    feat[(size_t)tok * 64 + 32 + lane] = (float)cnt;
    noise[base + lane] = nz;
    if (lane == 0) simlam[tok] = part + simacc + bf[0];
}

// ---------------- Kernel M: batched 64->32 tanh ->2 MLP via f32 WMMA -------
// One wave handles 16 tokens. D = A(16x4) x B(4x16) + C, 16 k-steps, 2 hidden tiles.
__global__ __launch_bounds__(32) void mlp_wmma_kernel(
    const float* __restrict__ feat,     // [TOK][64]
    const float* __restrict__ simlam,   // [TOK]
    const float* __restrict__ W2a,      // [32][64]
    const float* __restrict__ b2a,      // [32]
    const float* __restrict__ W2b,      // [2][32]
    const float* __restrict__ b2b,      // [2]
    float* __restrict__ lambda_out,     // d_out + TOK*VOC, [TOK]
    float* __restrict__ tempe)          // [TOK]
{
    __shared__ float Hs[16 * 33];
    const int lane = threadIdx.x;
    const int tile = blockIdx.x;
    const int half = lane >> 4;         // 0: K0/K1 lanes, 1: K2/K3 lanes
    const int l15  = lane & 15;

    const float* arow = feat + (size_t)(tile * 16 + l15) * 64 + half * 2;
    v8f d0 = {}; v8f d1 = {};
#pragma unroll
    for (int kk = 0; kk < 16; ++kk) {
        const int k0 = kk * 4;
        v2f a;  a.x  = arow[k0];  a.y  = arow[k0 + 1];
        const float* b0p = W2a + (size_t)l15 * 64 + k0 + half * 2;
        const float* b1p = W2a + (size_t)(l15 + 16) * 64 + k0 + half * 2;
        v2f b0; b0.x = b0p[0]; b0.y = b0p[1];
        v2f b1; b1.x = b1p[0]; b1.y = b1p[1];
        d0 = __builtin_amdgcn_wmma_f32_16x16x4_f32(false, a, false, b0, (short)0, d0, false, false);
        d1 = __builtin_amdgcn_wmma_f32_16x16x4_f32(false, a, false, b1, (short)0, d1, false, false);
    }

    // bias + tanh, stage H [16 tokens][32 hidden] in LDS (pad 33 vs 64 banks)
    const float bias0 = b2a[l15];
    const float bias1 = b2a[l15 + 16];
#pragma unroll
    for (int r = 0; r < 8; ++r) {
        int m = r + half * 8;                       // token row in tile (D layout)
        Hs[m * 33 + l15]      = tanhf(d0[r] + bias0);
        Hs[m * 33 + l15 + 16] = tanhf(d1[r] + bias1);
    }
    __syncthreads();

    // 32 lanes = 16 tokens x 2 outputs
    const int tt = lane >> 1, oo = lane & 1;
    float acc = b2b[oo];
#pragma unroll
    for (int h = 0; h < 32; ++h) acc += Hs[tt * 33 + h] * W2b[oo * 32 + h];
    float other = __shfl_xor(acc, 1);
    if (oo == 0) {
        int token = tile * 16 + tt;
        float sl = simlam[token];
        lambda_out[token] = 1.0f / (1.0f + expf(sl - acc));   // sigmoid(l0 - sim)
        tempe[token]      = 1.0f / (1.0f + expf(-other));     // sigmoid(l1)
    }
}

// ---------------- Kernel F: wave softmax over K + deterministic scatter ----
__global__ __launch_bounds__(256) void finalize_kernel(
    const int*   __restrict__ tgt,
    const float* __restrict__ dists,
    const float* __restrict__ noise,
    const float* __restrict__ tempe,
    float* __restrict__ out)            // knn_prob [TOK][VOC]
{
    const int lane = threadIdx.x & 31;
    const int tok  = blockIdx.x * 8 + (threadIdx.x >> 5);
    const size_t base = (size_t)tok * KNN;

    float logit = -dists[base + lane] * tempe[tok] + noise[base + lane];
    float m = logit;
#pragma unroll
    for (int off = 16; off >= 1; off >>= 1) m = fmaxf(m, __shfl_xor(m, off));
    float e = expf(logit - m);
    float s = e;
#pragma unroll
    for (int off = 16; off >= 1; off >>= 1) s += __shfl_xor(s, off);
    float p = e / s;

    int lbl = tgt[base + lane];
    bool dupA = false;                  // a later k has the same label -> it wins
    for (int j = 0; j < 32; ++j) {
        int lj = __shfl(lbl, j);
        if (j > lane && lj == lbl) dupA = true;
    }
    if (!dupA) out[(size_t)tok * VOC + lbl] = p;
}

extern "C" void kernel_launch(void* const* d_in, const int* in_sizes, int n_in,
                              void* d_out, int out_size, void* d_ws, size_t ws_size,
                              hipStream_t stream) {
    (void)in_sizes; (void)n_in; (void)out_size; (void)ws_size;
    const int*   tgt  = (const int*)  d_in[0];
    const float* dist = (const float*)d_in[1];
    const float* keyf = (const float*)d_in[2];
    const float* nprb = (const float*)d_in[3];
    const float* selp = (const float*)d_in[4];
    const float* Wf   = (const float*)d_in[5];
    const float* bf   = (const float*)d_in[6];
    const float* W1a  = (const float*)d_in[7];
    const float* b1a  = (const float*)d_in[8];
    const float* W1b  = (const float*)d_in[9];
    const float* b1b  = (const float*)d_in[10];
    const float* W2a  = (const float*)d_in[11];
    const float* b2a  = (const float*)d_in[12];
    const float* W2b  = (const float*)d_in[13];
    const float* b2b  = (const float*)d_in[14];

    float* out  = (float*)d_out;
    float* ws   = (float*)d_ws;
    float* feat  = ws;                        // TOK*64
    float* noise = feat + (size_t)TOK * 64;   // TOK*32
    float* siml  = noise + (size_t)TOK * 32;  // TOK
    float* temp  = siml + TOK;                // TOK

    zero_kernel<<<4096, 256, 0, stream>>>(out, (size_t)TOK * VOC / 4);
    token_kernel<<<TOK / 8, 256, 0, stream>>>(tgt, dist, keyf, nprb, selp,
                                              Wf, bf, W1a, b1a, W1b, b1b,
                                              feat, noise, siml);
    mlp_wmma_kernel<<<TOK / 16, 32, 0, stream>>>(feat, siml, W2a, b2a, W2b, b2b,
                                                 out + (size_t)TOK * VOC, temp);
    finalize_kernel<<<TOK / 8, 256, 0, stream>>>(tgt, dist, noise, temp, out);
}